// GraphConvolution_4776003633531
// MI455X (gfx1250) — compile-verified
//
#include <hip/hip_runtime.h>

typedef __attribute__((ext_vector_type(2))) float v2f;
typedef __attribute__((ext_vector_type(8))) float v8f;

#define FDIM 128                 // F_IN == CHANNELS == 128
#define LDS_STRIDE 132           // pad 128 -> 132 floats: kills 16-way bank conflicts

// ---------------------------------------------------------------------------
// Kernel 1: H = X @ W  using V_WMMA_F32_16X16X4_F32 (fp32 matrix path)
// grid.x = M/16 blocks, 256 threads = 8 waves; wave w owns 16-col tile w.
// ---------------------------------------------------------------------------
__global__ __launch_bounds__(256) void gemm_xw_wmma(
    const float* __restrict__ X, const float* __restrict__ W,
    float* __restrict__ H)
{
    __shared__ __align__(16) float As[16 * LDS_STRIDE];

    const int tid  = threadIdx.x;
    const int lane = tid & 31;
    const int wv   = tid >> 5;        // 0..7 : which 16-column tile of W/H
    const int mt   = blockIdx.x;      // 16-row tile of X/H

    // Cooperative load of the 16x128 A tile into LDS (contiguous in memory).
    const float* src = X + (size_t)mt * 16 * FDIM;
    for (int i = tid; i < 16 * (FDIM / 4); i += 256) {      // 512 float4 slots
        const int r  = i >> 5;        // row 0..15
        const int c4 = i & 31;        // float4 column 0..31
        *(float4*)&As[r * LDS_STRIDE + c4 * 4] =
            *(const float4*)(src + r * FDIM + c4 * 4);
    }
    __syncthreads();

    // WMMA f32 16x16x4 fragment mapping (ISA 7.12.2, 32-bit A 16x4):
    //   lanes 0-15: M=lane, K = {0,1}; lanes 16-31: M=lane-16, K = {2,3}
    const int mrow = lane & 15;
    const int kb   = (lane >> 4) * 2;
    const int n0   = wv * 16;

    v8f acc = {};
    #pragma unroll 4
    for (int k0 = 0; k0 < FDIM; k0 += 4) {
        v2f a, b;
        a.x = As[mrow * LDS_STRIDE + k0 + kb];
        a.y = As[mrow * LDS_STRIDE + k0 + kb + 1];
        b.x = W[(size_t)(k0 + kb)     * FDIM + n0 + mrow];
        b.y = W[(size_t)(k0 + kb + 1) * FDIM + n0 + mrow];
        // 8 args: (neg_a, A, neg_b, B, c_mod, C, reuse_a, reuse_b)
        acc = __builtin_amdgcn_wmma_f32_16x16x4_f32(
                  false, a, false, b, (short)0, acc, false, false);
    }

    // D layout: VGPR v -> row (v + 8*(lane>=16)), col = lane&15
    const int mbase = mt * 16 + ((lane >> 4) << 3);
    #pragma unroll
    for (int v = 0; v < 8; ++v) {
        H[(size_t)(mbase + v) * FDIM + n0 + mrow] = acc[v];
    }
}

// ---------------------------------------------------------------------------
// Kernel 2: out[n, c] = bias[c]   (vectorized float4; also clears the 0xAA
// poison so the scatter accumulates onto bias directly)
// ---------------------------------------------------------------------------
__global__ __launch_bounds__(256) void init_bias(
    float* __restrict__ out, const float* __restrict__ bias, int total4)
{
    const int i = blockIdx.x * 256 + threadIdx.x;
    if (i >= total4) return;
    const float4 b = ((const float4*)bias)[i & 31];   // 32 float4 per row of 128
    ((float4*)out)[i] = b;
}

// ---------------------------------------------------------------------------
// Kernel 3: scatter  out[row[e]] += w[e] * H[col[e]]
// One wave per edge: lane l handles channels 4l..4l+3 -> one coalesced 512B
// gather per edge (L2-resident: H is 51 MB < 192 MB L2), then 4 hardware
// fp32 atomics per lane.
// ---------------------------------------------------------------------------
__global__ __launch_bounds__(256) void edge_scatter(
    const int*   __restrict__ erow, const int* __restrict__ ecol,
    const float* __restrict__ ewgt, const float* __restrict__ H,
    float* __restrict__ out, int E)
{
    const int lane = threadIdx.x & 31;
    const int e    = blockIdx.x * 8 + (threadIdx.x >> 5);
    if (e >= E) return;

    const int   r = erow[e];
    const int   c = ecol[e];
    const float w = ewgt[e];

    const float4 hv = *(const float4*)(H + (size_t)c * FDIM + lane * 4);
    float* dst = out + (size_t)r * FDIM + lane * 4;

    unsafeAtomicAdd(dst + 0, w * hv.x);   // -> global_atomic_add_f32
    unsafeAtomicAdd(dst + 1, w * hv.y);
    unsafeAtomicAdd(dst + 2, w * hv.z);
    unsafeAtomicAdd(dst + 3, w * hv.w);
}

// ---------------------------------------------------------------------------
// Kernel 4: in-place ReLU (vectorized)
// ---------------------------------------------------------------------------
__global__ __launch_bounds__(256) void relu_inplace(float* __restrict__ out,
                                                    int total4)
{
    const int i = blockIdx.x * 256 + threadIdx.x;
    if (i >= total4) return;
    float4 v = ((float4*)out)[i];
    v.x = fmaxf(v.x, 0.0f);
    v.y = fmaxf(v.y, 0.0f);
    v.z = fmaxf(v.z, 0.0f);
    v.w = fmaxf(v.w, 0.0f);
    ((float4*)out)[i] = v;
}

// ---------------------------------------------------------------------------
// inputs: x[100000*128] f32, edge_row[1.6M] i32, edge_col[1.6M] i32,
//         edge_weight[1.6M] f32, kernel[128*128] f32, bias[128] f32
// d_ws: H = X@W scratch (100000*128*4 = 51.2 MB)
// ---------------------------------------------------------------------------
extern "C" void kernel_launch(void* const* d_in, const int* in_sizes, int n_in,
                              void* d_out, int out_size, void* d_ws, size_t ws_size,
                              hipStream_t stream) {
    const float* X    = (const float*)d_in[0];
    const int*   erow = (const int*)  d_in[1];
    const int*   ecol = (const int*)  d_in[2];
    const float* ewgt = (const float*)d_in[3];
    const float* W    = (const float*)d_in[4];
    const float* bias = (const float*)d_in[5];
    float* out = (float*)d_out;
    float* H   = (float*)d_ws;

    const int M = in_sizes[0] / FDIM;   // 100000 (divisible by 16)
    const int E = in_sizes[1];          // 1,600,000 (divisible by 8)
    const int total4 = M * (FDIM / 4);  // float4 count of output

    gemm_xw_wmma<<<M / 16, 256, 0, stream>>>(X, W, H);
    init_bias<<<(total4 + 255) / 256, 256, 0, stream>>>(out, bias, total4);
    edge_scatter<<<(E + 7) / 8, 256, 0, stream>>>(erow, ecol, ewgt, H, out, E);
    relu_inplace<<<(total4 + 255) / 256, 256, 0, stream>>>(out, total4);
}